// FCNNShapeCounterValuationFunction_27419071217674
// MI455X (gfx1250) — compile-verified
//
#include <hip/hip_runtime.h>

typedef __attribute__((ext_vector_type(2))) float v2f;
typedef __attribute__((ext_vector_type(8))) float v8f;

#define SCALE 0.999f

// One 16-row tile per wave32: D = A(16x16) x S^T(16x16) via 4x V_WMMA_F32_16X16X4_F32,
// where S^T[k][j] = SCALE * (idx[row0+j] == k). Diagonal of D is the output.
__global__ __launch_bounds__(256)
void onehot_gather_wmma(const float* __restrict__ z,
                        const float* __restrict__ a,
                        const int*   __restrict__ attr_ptr,
                        float*       __restrict__ out,
                        int zstride, long n_tiles)
{
    const int  lane = threadIdx.x & 31;
    const long wave = ((long)blockIdx.x * (long)blockDim.x + (long)threadIdx.x) >> 5;
    if (wave >= n_tiles) return;            // wave-uniform: EXEC stays all-1s for live waves

    const int  attr = attr_ptr[0];
    const long row0 = wave * 16;
    const int  m    = lane & 15;            // matrix row (A) / column (B,S^T) this lane owns
    const long r    = row0 + m;

    // one-hot index for column m of S^T (both lane halves need it; L2/L0 absorb the dup)
    const int idxm = (int)__builtin_nontemporal_load(&z[r * (long)zstride + attr]);

    // A-matrix 16x4 f32 layout per chunk c (K = 4c..4c+3):
    //   lanes 0-15:  VGPR0 = K=4c,   VGPR1 = K=4c+1
    //   lanes 16-31: VGPR0 = K=4c+2, VGPR1 = K=4c+3
    const int  koff = (lane < 16) ? 0 : 2;  // K offset within the chunk for this lane half
    const int  h    = koff >> 1;            // float2 index offset
    const v2f* arow = (const v2f*)(a + r * 16);
    v2f A0 = __builtin_nontemporal_load(arow + 0 + h);
    v2f A1 = __builtin_nontemporal_load(arow + 2 + h);
    v2f A2 = __builtin_nontemporal_load(arow + 4 + h);
    v2f A3 = __builtin_nontemporal_load(arow + 6 + h);

    // B-matrix 4x16 f32 (mirror of A): lane holds column N=m, VGPRs hold rows K.
    // B chunk c, VGPR0 -> K = 4c+koff, VGPR1 -> K = 4c+koff+1.
    v2f B0 = { (idxm == 0  + koff) ? SCALE : 0.0f, (idxm == 1  + koff) ? SCALE : 0.0f };
    v2f B1 = { (idxm == 4  + koff) ? SCALE : 0.0f, (idxm == 5  + koff) ? SCALE : 0.0f };
    v2f B2 = { (idxm == 8  + koff) ? SCALE : 0.0f, (idxm == 9  + koff) ? SCALE : 0.0f };
    v2f B3 = { (idxm == 12 + koff) ? SCALE : 0.0f, (idxm == 13 + koff) ? SCALE : 0.0f };

    v8f c8 = {};
    c8 = __builtin_amdgcn_wmma_f32_16x16x4_f32(false, A0, false, B0, (short)0, c8, false, false);
    c8 = __builtin_amdgcn_wmma_f32_16x16x4_f32(false, A1, false, B1, (short)0, c8, false, false);
    c8 = __builtin_amdgcn_wmma_f32_16x16x4_f32(false, A2, false, B2, (short)0, c8, false, false);
    c8 = __builtin_amdgcn_wmma_f32_16x16x4_f32(false, A3, false, B3, (short)0, c8, false, false);

    // Diagonal extraction. C/D 16x16 f32 layout:
    //   lanes 0-15:  VGPR v holds D[v][lane]      -> D[i][i] at lane i,    VGPR i   (i<8)
    //   lanes 16-31: VGPR v holds D[v+8][lane-16] -> D[i][i] at lane i+16, VGPR i-8 (i>=8)
    const int sel = lane & 7;
    float t0 = (sel & 1) ? c8[1] : c8[0];
    float t1 = (sel & 1) ? c8[3] : c8[2];
    float t2 = (sel & 1) ? c8[5] : c8[4];
    float t3 = (sel & 1) ? c8[7] : c8[6];
    float u0 = (sel & 2) ? t1 : t0;
    float u1 = (sel & 2) ? t3 : t2;
    float v  = (sel & 4) ? u1 : u0;

    const bool active = (lane < 8) || (lane >= 24);
    const long i = row0 + ((lane < 8) ? (long)lane : (long)(lane - 16));
    if (active) __builtin_nontemporal_store(v, &out[i]);
}

// Scalar fallback / tail: out[i] = SCALE * a[i*K + (int)z[i*D + attr]]
__global__ __launch_bounds__(256)
void onehot_gather_scalar(const float* __restrict__ z,
                          const float* __restrict__ a,
                          const int*   __restrict__ attr_ptr,
                          float*       __restrict__ out,
                          int zstride, int astride, long start, long n)
{
    const long i = start + (long)blockIdx.x * (long)blockDim.x + (long)threadIdx.x;
    if (i < n) {
        const int idx = (int)__builtin_nontemporal_load(&z[i * (long)zstride + attr_ptr[0]]);
        const float val = __builtin_nontemporal_load(&a[i * (long)astride + idx]);
        __builtin_nontemporal_store(SCALE * val, &out[i]);
    }
}

extern "C" void kernel_launch(void* const* d_in, const int* in_sizes, int n_in,
                              void* d_out, int out_size, void* d_ws, size_t ws_size,
                              hipStream_t stream)
{
    const float* z    = (const float*)d_in[0];
    const float* a    = (const float*)d_in[1];
    const int*   attr = (const int*)d_in[2];
    float*       out  = (float*)d_out;

    const long n = (long)out_size;                 // B rows
    if (n <= 0) return;
    const int D = (int)((long)in_sizes[0] / n);    // z row stride (32)
    const int K = (int)((long)in_sizes[1] / n);    // a row stride (16)

    const int threads = 256;

    if (K == 16) {
        const long tiles = n / 16;
        const long rem   = n - tiles * 16;
        if (tiles > 0) {
            const long waves_per_block = threads / 32;
            const long blocks = (tiles + waves_per_block - 1) / waves_per_block;
            onehot_gather_wmma<<<(unsigned)blocks, threads, 0, stream>>>(
                z, a, attr, out, D, tiles);
        }
        if (rem > 0) {
            onehot_gather_scalar<<<1, threads, 0, stream>>>(
                z, a, attr, out, D, K, n - rem, n);
        }
    } else {
        const long blocks = (n + threads - 1) / threads;
        onehot_gather_scalar<<<(unsigned)blocks, threads, 0, stream>>>(
            z, a, attr, out, D, K, 0, n);
    }
}